// TrainNet_12386685682456
// MI455X (gfx1250) — compile-verified
//
#include <hip/hip_runtime.h>
#include <hip/hip_bf16.h>

typedef __attribute__((ext_vector_type(2))) float v2f;
typedef __attribute__((ext_vector_type(8))) float v8f;

#define FEAT 512  // feature width of x and h (aggregation width, GEMM K)

// ---------------------------------------------------------------------------
// z[i] = src[i]   (vectorized float4 copy; initializes the GIN self-term)
// ---------------------------------------------------------------------------
__global__ void copy_f32_kernel(const float* __restrict__ in,
                                float* __restrict__ out, long long n4) {
    long long i = (long long)blockIdx.x * blockDim.x + threadIdx.x;
    if (i < n4) {
        ((float4*)out)[i] = ((const float4*)in)[i];
    }
}

// ---------------------------------------------------------------------------
// z[dst[e]] += feat[src[e]]  via hardware global_atomic_add_f32.
// One 256-thread block per edge; 2 scalar f32 atomics per thread (512 feats).
// ---------------------------------------------------------------------------
__global__ void scatter_add_kernel(const float* __restrict__ feat,
                                   float* __restrict__ z,
                                   const int* __restrict__ src_idx,
                                   const int* __restrict__ dst_idx,
                                   int nedges) {
    int e = blockIdx.x;
    if (e >= nedges) return;
    int s = src_idx[e];
    int d = dst_idx[e];
    const float* fs = feat + (size_t)s * FEAT;
    float*       fd = z    + (size_t)d * FEAT;
    int t = threadIdx.x;              // 0..255
    float a0 = fs[t];
    float a1 = fs[t + 256];
    unsafeAtomicAdd(&fd[t],       a0);   // -> global_atomic_add_f32
    unsafeAtomicAdd(&fd[t + 256], a1);
}

// ---------------------------------------------------------------------------
// out[M][N] = (RELU?)(A[M][512] @ W[512][N] + bias[N])   with N compile-time.
// wave32: each wave computes a 32x64 strip with V_WMMA_F32_16X16X4_F32.
//   A fragment (16x4 f32): lanes 0-15 -> K=k0,k0+1 ; lanes 16-31 -> k0+2,k0+3
//   B fragment (4x16 f32): VGPR0 = row (k0+2*khalf), VGPR1 = row (k0+2*khalf+1)
//   C/D (16x16 f32, 8 VGPRs): VGPR v -> rows v (lanes 0-15) and v+8 (16-31)
// Two M fragments per wave -> each B fragment feeds 2 WMMAs (1 load / WMMA).
// ---------------------------------------------------------------------------
#define WMMA_F32(a, b, c) \
    __builtin_amdgcn_wmma_f32_16x16x4_f32(false, (a), false, (b), (short)0, (c), false, false)

template <int RELU, int N>
__global__ __launch_bounds__(256)
void gemm_wmma_f32_kernel(const float* __restrict__ A,     // [M][512]
                          const float* __restrict__ W,     // [512][N]
                          const float* __restrict__ bias,  // [N]
                          float* __restrict__ out,         // [M][N]
                          int M) {
    constexpr int K = FEAT;                      // 512
    constexpr int KPF = 16;                      // prefetch distance (K rows)
    const int wave  = threadIdx.x >> 5;          // 0..7
    const int lane  = threadIdx.x & 31;
    const int mBase = (blockIdx.x * 8 + wave) * 32;
    if (mBase >= M) return;                      // wave-uniform: EXEC stays full
    const bool hiValid = (mBase + 16) < M;       // wave-uniform (M % 16 == 0)
    const int nBase = blockIdx.y * 64;

    const int row   = lane & 15;
    const int khalf = lane >> 4;                 // 0 or 1
    const int col   = nBase + row;               // this lane's N column (per tile +16)

    v8f c00 = {}, c01 = {}, c02 = {}, c03 = {};  // rows mBase..mBase+15
    v8f c10 = {}, c11 = {}, c12 = {}, c13 = {};  // rows mBase+16..mBase+31

    const int r0 = mBase + row;
    const int r1 = hiValid ? (r0 + 16) : r0;     // clamp keeps loads in-range
    const float* aRow0 = A + (size_t)r0 * K + 2 * khalf;
    const float* aRow1 = A + (size_t)r1 * K + 2 * khalf;

#define GEMM_K_STEP(k0, DO_PREFETCH)                                          \
    {                                                                         \
        v2f a0 = *(const v2f*)(aRow0 + (k0));                                 \
        v2f a1 = *(const v2f*)(aRow1 + (k0));                                 \
        const float* w0 = W + (size_t)((k0) + 2 * khalf) * N;                 \
        const float* w1 = w0 + N;                                            \
        v2f b0, b1, b2, b3;                                                   \
        b0[0] = w0[col];      b0[1] = w1[col];                                \
        b1[0] = w0[col + 16]; b1[1] = w1[col + 16];                           \
        b2[0] = w0[col + 32]; b2[1] = w1[col + 32];                           \
        b3[0] = w0[col + 48]; b3[1] = w1[col + 48];                           \
        if (DO_PREFETCH) {                                                    \
            /* WGP-scope prefetch (locality 3): fill all cache levels */      \
            __builtin_prefetch(w0 + (size_t)KPF * N + col, 0, 3);             \
        }                                                                     \
        c00 = WMMA_F32(a0, b0, c00);                                          \
        c10 = WMMA_F32(a1, b0, c10);                                          \
        c01 = WMMA_F32(a0, b1, c01);                                          \
        c11 = WMMA_F32(a1, b1, c11);                                          \
        c02 = WMMA_F32(a0, b2, c02);                                          \
        c12 = WMMA_F32(a1, b2, c12);                                          \
        c03 = WMMA_F32(a0, b3, c03);                                          \
        c13 = WMMA_F32(a1, b3, c13);                                          \
    }

    // Main loop (with prefetch), then short tail without it: no branch in loop
    for (int k0 = 0; k0 < K - KPF; k0 += 4) {
        GEMM_K_STEP(k0, 1)
    }
    for (int k0 = K - KPF; k0 < K; k0 += 4) {
        GEMM_K_STEP(k0, 0)
    }
#undef GEMM_K_STEP

    // Epilogue: bias (+ReLU), store per documented C/D layout
    const float bv0 = bias[col];
    const float bv1 = bias[col + 16];
    const float bv2 = bias[col + 32];
    const float bv3 = bias[col + 48];

#pragma unroll
    for (int v = 0; v < 8; ++v) {
        const int r = mBase + v + khalf * 8;
        float* o = out + (size_t)r * N + col;
        float x0 = c00[v] + bv0;
        float x1 = c01[v] + bv1;
        float x2 = c02[v] + bv2;
        float x3 = c03[v] + bv3;
        if (RELU) {
            x0 = fmaxf(x0, 0.0f); x1 = fmaxf(x1, 0.0f);
            x2 = fmaxf(x2, 0.0f); x3 = fmaxf(x3, 0.0f);
        }
        o[0] = x0; o[16] = x1; o[32] = x2; o[48] = x3;
    }

    if (hiValid) {
#pragma unroll
        for (int v = 0; v < 8; ++v) {
            const int r = mBase + 16 + v + khalf * 8;
            float* o = out + (size_t)r * N + col;
            float x0 = c10[v] + bv0;
            float x1 = c11[v] + bv1;
            float x2 = c12[v] + bv2;
            float x3 = c13[v] + bv3;
            if (RELU) {
                x0 = fmaxf(x0, 0.0f); x1 = fmaxf(x1, 0.0f);
                x2 = fmaxf(x2, 0.0f); x3 = fmaxf(x3, 0.0f);
            }
            o[0] = x0; o[16] = x1; o[32] = x2; o[48] = x3;
        }
    }
}

// ---------------------------------------------------------------------------
// Orchestration: z1 = x + scatter(x); h = relu(z1@W1+b1);
//                z2 = h + scatter(h); out = z2@W2+b2
// ---------------------------------------------------------------------------
extern "C" void kernel_launch(void* const* d_in, const int* in_sizes, int n_in,
                              void* d_out, int out_size, void* d_ws, size_t ws_size,
                              hipStream_t stream) {
    const float* x    = (const float*)d_in[0];
    const int*   edge = (const int*)d_in[1];   // [2][E] flattened
    const float* W1   = (const float*)d_in[2];
    const float* b1   = (const float*)d_in[3];
    const float* W2   = (const float*)d_in[4];
    const float* b2   = (const float*)d_in[5];

    const int M  = in_sizes[0] / FEAT;   // 50000 nodes
    const int E  = in_sizes[1] / 2;      // 400000 edges

    const int* srcI = edge;
    const int* dstI = edge + E;

    float* z = (float*)d_ws;                 // [M][512] aggregate buffer
    float* h = z + (size_t)M * FEAT;         // [M][512] hidden activations

    const long long n4 = (long long)M * FEAT / 4;
    const int copyBlocks = (int)((n4 + 255) / 256);
    const int mTiles32 = (M + 31) / 32;
    const int gx = (mTiles32 + 7) / 8;       // 8 waves (32 rows each) per block

    // ---- layer 1 ----
    copy_f32_kernel<<<copyBlocks, 256, 0, stream>>>(x, z, n4);
    scatter_add_kernel<<<E, 256, 0, stream>>>(x, z, srcI, dstI, E);
    gemm_wmma_f32_kernel<1, 512><<<dim3(gx, 512 / 64), 256, 0, stream>>>(z, W1, b1, h, M);

    // ---- layer 2 ----
    copy_f32_kernel<<<copyBlocks, 256, 0, stream>>>(h, z, n4);
    scatter_add_kernel<<<E, 256, 0, stream>>>(h, z, srcI, dstI, E);
    gemm_wmma_f32_kernel<0, 256><<<dim3(gx, 256 / 64), 256, 0, stream>>>(z, W2, b2, (float*)d_out, M);
}